// DirectDepthMapper_39281770889514
// MI455X (gfx1250) — compile-verified
//
#include <hip/hip_runtime.h>
#include <hip/hip_bf16.h>

#define IMG_W 4096
#define IMG_H 4096
#define TILE_ROWS 4
#define TILE_ELEMS (TILE_ROWS * IMG_W)     // 16384 floats
#define TILE_BYTES (TILE_ELEMS * 4)        // 65536 bytes
#define GRID_N 400
#define NBINS (GRID_N * GRID_N)            // 160000
#define MAX_COPIES 16

typedef __attribute__((ext_vector_type(4))) unsigned int uint4v;
typedef __attribute__((ext_vector_type(8))) int int8v;
typedef __attribute__((ext_vector_type(4))) int int4v;

// Issue one TDM load: tile of TILE_ROWS x IMG_W f32 from global -> LDS offset ldsOff.
__device__ __forceinline__ void tdm_issue_tile(const float* depth, int row0, unsigned ldsOff) {
    unsigned long long ga = (unsigned long long)(size_t)depth
                          + (unsigned long long)row0 * (IMG_W * 4ull);
    // D# group 0 (128b): [1:0]=count=1, [63:32]=lds_addr, [120:64]=global_addr, [127:126]=type=2
    uint4v g0;
    g0.x = 1u;
    g0.y = ldsOff;
    g0.z = (unsigned)(ga & 0xFFFFFFFFull);
    g0.w = (unsigned)((ga >> 32) & 0x1FFFFFFull) | (2u << 30);
    // D# group 1 (256b)
    int8v g1;
    g1[0] = 0x20000;                               // data_size=2 (4 bytes), mask=0, no pad/iterate
    g1[1] = (int)(((unsigned)IMG_W & 0xFFFFu) << 16);       // tensor_dim0[15:0] at [63:48]
    g1[2] = (int)((((unsigned)IMG_W >> 16) & 0xFFFFu)       // tensor_dim0[31:16]
                  | (((unsigned)IMG_H & 0xFFFFu) << 16));   // tensor_dim1[15:0]
    g1[3] = (int)((((unsigned)IMG_H >> 16) & 0xFFFFu)       // tensor_dim1[31:16]
                  | (((unsigned)IMG_W & 0xFFFFu) << 16));   // tile_dim0 = 4096
    g1[4] = TILE_ROWS;                             // tile_dim1 = 4, tile_dim2 = 0
    g1[5] = IMG_W;                                 // tensor_dim0_stride[31:0]
    g1[6] = 0;                                     // stride0 hi / stride1 lo
    g1[7] = 0;
    int4v z4 = {0, 0, 0, 0};                       // groups 2/3 unused (2D tensor)
    int8v z8 = {0, 0, 0, 0, 0, 0, 0, 0};
    __builtin_amdgcn_tensor_load_to_lds(g0, g1, z4, z4, z8, 0);
}

__global__ void zero_u32_kernel(unsigned* __restrict__ p, int n) {
    int i = blockIdx.x * blockDim.x + threadIdx.x;
    if (i < n) p[i] = 0u;
}

__global__ void depth_map_kernel(const float* __restrict__ depth,
                                 const float* __restrict__ pose,
                                 unsigned* __restrict__ acc, int K) {
    extern __shared__ __align__(16) float smem[];   // TILE_BYTES dynamic LDS

    const int tile = blockIdx.x;           // one tile (4 rows) per block
    const int row0 = tile * TILE_ROWS;

    if (threadIdx.x == 0) {
        unsigned ldsOff = (unsigned)(size_t)&smem[0];   // low 32 bits = LDS byte offset
        tdm_issue_tile(depth, row0, ldsOff);
        __builtin_amdgcn_s_wait_tensorcnt(0);
    }
    __syncthreads();

    // Pose rows (uniform -> scalar loads)
    const float p00 = pose[0],  p01 = pose[1],  p02 = pose[2],  p03 = pose[3];
    const float p10 = pose[4],  p11 = pose[5],  p12 = pose[6],  p13 = pose[7];
    const float p20 = pose[8],  p21 = pose[9],  p22 = pose[10], p23 = pose[11];

    unsigned* __restrict__ bins = acc + (size_t)(tile % K) * NBINS;

    const float4* sb = (const float4*)smem;
    for (int q = threadIdx.x; q < TILE_ELEMS / 4; q += blockDim.x) {
        float4 d4 = sb[q];                  // ds_load_b128
        int pb = q * 4;
        int r  = pb >> 12;                  // row within tile (IMG_W = 4096)
        int c  = pb & (IMG_W - 1);
        int j  = row0 + r;
        float ym = (float)(j - 2047);       // cy = 2047
        float dv[4] = {d4.x, d4.y, d4.z, d4.w};
#pragma unroll
        for (int e = 0; e < 4; ++e) {
            float d  = dv[e];
            float xm = (float)(c + e - 2047);            // cx = 2047
            float X  = d * xm * (1.0f / 4096.0f);        // /fx
            float Y  = d * ym * (1.0f / 4096.0f);        // /fy
            float Z  = d;
            float gx =  fmaf(p00, X, fmaf(p01, Y, fmaf(p02, Z, p03)));
            float gy = -fmaf(p10, X, fmaf(p11, Y, fmaf(p12, Z, p13)));  // + CAMERA_HEIGHT(0)
            float gz =  fmaf(p20, X, fmaf(p21, Y, fmaf(p22, Z, p23)));
            bool ok = (fabsf(Z) < 4.0f) && (fabsf(Z) >= 0.1f) &&
                      (gy > 0.0f) && (gy < 1.0f);
            float fz = rintf(gz / 0.1f + 200.0f);        // round-half-even like jnp.round
            float fx = rintf(gx / 0.1f + 200.0f);
            ok = ok && (fz >= 0.0f) && (fz <= (float)(GRID_N - 1)) &&
                       (fx >= 0.0f) && (fx <= (float)(GRID_N - 1));
            if (ok) {
                int bin = (int)fz * GRID_N + (int)fx;
                atomicAdd(&bins[bin], 1u);               // no-return global_atomic_add_u32
            }
        }
    }
}

__global__ void reduce_kernel(const unsigned* __restrict__ acc, float* __restrict__ out, int K) {
    int i = blockIdx.x * blockDim.x + threadIdx.x;
    if (i < NBINS) {
        unsigned s = 0;
        for (int k = 0; k < K; ++k) s += acc[(size_t)k * NBINS + i];
        out[i] = (float)s;
    }
}

extern "C" void kernel_launch(void* const* d_in, const int* in_sizes, int n_in,
                              void* d_out, int out_size, void* d_ws, size_t ws_size,
                              hipStream_t stream) {
    const float* depth = (const float*)d_in[0];
    const float* pose  = (const float*)d_in[1];
    float* out = (float*)d_out;

    const size_t gridBytes = (size_t)NBINS * 4;
    int K = (int)(ws_size / gridBytes);
    if (K > MAX_COPIES) K = MAX_COPIES;

    unsigned* acc;
    if (K >= 1) {
        acc = (unsigned*)d_ws;             // K privatized u32 histograms in workspace
    } else {
        K = 1;
        acc = (unsigned*)d_out;            // fallback: accumulate u32 in-place, convert in reduce
    }

    int zn = K * NBINS;
    zero_u32_kernel<<<(zn + 255) / 256, 256, 0, stream>>>(acc, zn);

    depth_map_kernel<<<IMG_H / TILE_ROWS, 256, TILE_BYTES, stream>>>(depth, pose, acc, K);

    reduce_kernel<<<(NBINS + 255) / 256, 256, 0, stream>>>(acc, out, K);
}